// LIFSpike_83416854823339
// MI455X (gfx1250) — compile-verified
//
#include <hip/hip_runtime.h>
#include <stdint.h>

// ---------------------------------------------------------------------------
// LIF spiking-neuron scan, CDNA5 (gfx1250) — Tensor Data Mover edition.
//
// x:   [B=32, C=128, T=16, H=32, W=32] fp32  (T-stride = H*W = 1024 floats)
// out: spikes, same shape/dtype. Pure HBM-bandwidth problem (~23 us floor).
//
//   * one block per (b,c) slice
//   * ONE tensor_load_to_lds DMA per block stages the 16x4096B tile
//     (2D D# descriptor, TENSORcnt) -> LDS row-major [t][hw], so per-t
//     reads are 128 contiguous dwords per wave: bank-conflict free
//   * sequential scan in registers, nontemporal b128 stores for spikes
// ---------------------------------------------------------------------------

typedef float f32x4 __attribute__((ext_vector_type(4)));
typedef int   i32x4 __attribute__((ext_vector_type(4)));
typedef int   i32x8 __attribute__((ext_vector_type(8)));

#define T_STEPS 16
#define HW      1024               // H*W floats per (o,t) plane
#define COLS    256                // float4 columns per plane = HW/4
#define THRESH  0.5f
#define BETA    0.75f

__global__ __launch_bounds__(256) void lif_scan_kernel(
    const float* __restrict__ x, float* __restrict__ out, int nOuter)
{
    // 16 rows x 4096 B = 64 KB, row-major [t][hw4]
    __shared__ f32x4 tile[T_STEPS * COLS];

    const int tid = threadIdx.x;       // float4 column 0..255
    const int o   = blockIdx.x;        // (b*C + c) slice index (block-uniform)
    if (o >= nOuter) return;

    // ---- wave 0 issues one TDM descriptor for the whole 64 KB tile ----
    if (tid < 32) {
        uint64_t ga  = (uint64_t)(uintptr_t)x + (uint64_t)o * (T_STEPS * HW * 4);
        uint32_t lds = (uint32_t)(uintptr_t)(&tile[0]);

        // D# group 0 (4 SGPRs): count=1 | lds_addr | global_addr[56:0] | type=2
        i32x4 d0;
        d0[0] = 1;                                    // count=1, user desc
        d0[1] = (int)lds;                             // lds_addr (bytes)
        d0[2] = (int)(uint32_t)ga;                    // global_addr[31:0]
        d0[3] = (int)((uint32_t)((ga >> 32) & 0x01FFFFFFu) | 0x80000000u); // [56:32]|type=2

        // D# group 1 (8 SGPRs)
        i32x8 d1;
        d1[0] = 0x00020000;                 // wg_mask=0, data_size=2 (4B), no pad/iter
        d1[1] = (HW & 0xFFFF) << 16;        // [47:32]=0, tensor_dim0[15:0]=1024
        d1[2] = T_STEPS << 16;              // tensor_dim0[31:16]=0, tensor_dim1=16
        d1[3] = HW << 16;                   // tensor_dim1 hi=0, tile_dim0=1024
        d1[4] = T_STEPS;                    // tile_dim1=16, tile_dim2=0
        d1[5] = HW;                         // tensor_dim0_stride[31:0]=1024
        d1[6] = 0;                          // stride hi=0, tensor_dim1_stride lo=0
        d1[7] = 0;                          // tensor_dim1_stride hi=0

        asm volatile("tensor_load_to_lds %0, %1"
                     :
                     : "s"(d0), "s"(d1)
                     : "memory");
        __builtin_amdgcn_s_wait_tensorcnt(0);
    }
    __syncthreads();   // publish DMA'd tile to all 8 waves

    // ---- sequential LIF scan over T ----
    float* op = out + (size_t)o * (T_STEPS * HW) + (size_t)tid * 4;

    f32x4 mem = {0.0f, 0.0f, 0.0f, 0.0f};
#pragma unroll
    for (int t = 0; t < T_STEPS; ++t) {
        f32x4 xv = tile[t * COLS + tid];   // ds_load_b128, conflict-free
        mem = mem * BETA + xv;             // v_fma
        f32x4 s;
        s.x = (mem.x > THRESH) ? 1.0f : 0.0f;
        s.y = (mem.y > THRESH) ? 1.0f : 0.0f;
        s.z = (mem.z > THRESH) ? 1.0f : 0.0f;
        s.w = (mem.w > THRESH) ? 1.0f : 0.0f;
        mem = mem - s * THRESH;            // soft reset
        __builtin_nontemporal_store(s, (f32x4*)(op + t * HW));
    }
}

extern "C" void kernel_launch(void* const* d_in, const int* in_sizes, int n_in,
                              void* d_out, int out_size, void* d_ws, size_t ws_size,
                              hipStream_t stream) {
    const float* x   = (const float*)d_in[0];
    float*       out = (float*)d_out;

    int nOuter = in_sizes[0] / (T_STEPS * HW);   // 32*128 = 4096 slices

    dim3 grid((unsigned)nOuter);
    dim3 block(256);
    hipLaunchKernelGGL(lif_scan_kernel, grid, block, 0, stream, x, out, nOuter);
}